// StateSpaceLayer_69449621176587
// MI455X (gfx1250) — compile-verified
//
#include <hip/hip_runtime.h>
#include <cstdint>

// Problem geometry (from the reference): x is (B, T, D) f32.
#define TLEN 4096
#define DIM  1024
#define CH   16                 // channels per block
#define KCH  64                 // chunks along T per block
#define LCH  (TLEN / KCH)       // 64 steps per chunk
#define NTH  1024               // threads per block = CH * KCH
// LDS tile layout: [chunk][i][c], with 16-float pad per chunk so the two
// half-wave lane groups (k and k+1) land on disjoint LDS banks.
#define CSF   (LCH * CH + 16)   // chunk stride in floats = 1040
#define XTILE (KCH * CSF)       // 66560 floats = 266240 B

__global__ void __launch_bounds__(NTH)
ssm_scan_69449621176587(const float* __restrict__ x,
                        const float* __restrict__ logit_a,
                        const float* __restrict__ bcoef,
                        const float* __restrict__ ccoef,
                        const float* __restrict__ dcoef,
                        float* __restrict__ out) {
    extern __shared__ float smem[];
    float* xs = smem;                       // XTILE floats (x tile, then y in-place)
    float* pP = smem + XTILE;               // KCH*CH chunk partial states
    float* pE = pP + KCH * CH;              // KCH*CH chunk entry states

    const int tid = threadIdx.x;
    const int blk = blockIdx.x;
    const int bb  = blk >> 6;               // 64 channel-groups per batch
    const int d0  = (blk & 63) * CH;

    const float* gx = x   + (size_t)bb * TLEN * DIM + d0;
    float*       gy = out + (size_t)bb * TLEN * DIM + d0;

    // Per-thread channel coefficients (issued early, overlap with async copies).
    const int c   = tid & (CH - 1);
    const int k   = tid >> 4;
    const int dch = d0 + c;
    const float a  = tanhf(logit_a[dch]);
    const float bc = bcoef[dch];
    const float cc = ccoef[dch];
    const float dc = dcoef[dch];
    float aL = a;                            // a^64 via 6 squarings (sign-correct)
#pragma unroll
    for (int q = 0; q < 6; ++q) aL *= aL;

    // ---- Stage in: async global -> LDS, float4 granularity -------------------
    // linear float4 index v = j*NTH + tid over TLEN*CH/4 = 16384 vectors.
#pragma unroll
    for (int j = 0; j < (TLEN * CH / 4) / NTH; ++j) {
        const int v  = j * NTH + tid;
        const int t  = v >> 2;               // time row
        const int c4 = v & 3;                // which float4 of the 16-ch row
        const uint64_t gaddr =
            (uint64_t)(uintptr_t)(gx + (size_t)t * DIM + c4 * 4);
        const uint32_t laddr = (uint32_t)(uintptr_t)
            (xs + (t >> 6) * CSF + (t & 63) * CH + c4 * 4);
        asm volatile("global_load_async_to_lds_b128 %0, %1, off"
                     :: "v"(laddr), "v"(gaddr) : "memory");
    }
    asm volatile("s_wait_asynccnt 0" ::: "memory");
    __syncthreads();

    // ---- Pass 1: chunk-local scan with zero entry state ----------------------
    {
        const float* xc = xs + k * CSF + c;
        float s = 0.0f;
#pragma unroll 8
        for (int i = 0; i < LCH; ++i)
            s = fmaf(a, s, bc * xc[i * CH]);
        pP[k * CH + c] = s;
    }
    __syncthreads();

    // ---- Cross-chunk serial scan (16 lanes, 64 steps, all in LDS) ------------
    if (tid < CH) {
        float s = 0.0f;
        for (int kk = 0; kk < KCH; ++kk) {
            pE[kk * CH + tid] = s;
            s = fmaf(aL, s, pP[kk * CH + tid]);
        }
    }
    __syncthreads();

    // ---- Pass 2: exact scan from entry state; write y over x in place --------
    {
        float* xc = xs + k * CSF + c;
        float s = pE[k * CH + c];
#pragma unroll 8
        for (int i = 0; i < LCH; ++i) {
            const float u = xc[i * CH];
            s = fmaf(a, s, bc * u);
            xc[i * CH] = fmaf(cc, s, dc * u);
        }
    }
    __syncthreads();   // all LDS writes visible before async store-out

    // ---- Stage out: async LDS -> global --------------------------------------
#pragma unroll
    for (int j = 0; j < (TLEN * CH / 4) / NTH; ++j) {
        const int v  = j * NTH + tid;
        const int t  = v >> 2;
        const int c4 = v & 3;
        const uint64_t gaddr =
            (uint64_t)(uintptr_t)(gy + (size_t)t * DIM + c4 * 4);
        const uint32_t laddr = (uint32_t)(uintptr_t)
            (xs + (t >> 6) * CSF + (t & 63) * CH + c4 * 4);
        asm volatile("global_store_async_from_lds_b128 %0, %1, off"
                     :: "v"(gaddr), "v"(laddr) : "memory");
    }
    asm volatile("s_wait_asynccnt 0" ::: "memory");
}

extern "C" void kernel_launch(void* const* d_in, const int* in_sizes, int n_in,
                              void* d_out, int out_size, void* d_ws, size_t ws_size,
                              hipStream_t stream) {
    const float* x  = (const float*)d_in[0];
    const float* la = (const float*)d_in[1];
    const float* b  = (const float*)d_in[2];
    const float* c  = (const float*)d_in[3];
    const float* d  = (const float*)d_in[4];
    float* out = (float*)d_out;

    const int batches = in_sizes[0] / (TLEN * DIM);          // = 8
    const int blocks  = batches * (DIM / CH);                // = 512
    const size_t shmem = (size_t)(XTILE + 2 * KCH * CH) * sizeof(float); // ~268 KB

    hipLaunchKernelGGL(ssm_scan_69449621176587,
                       dim3(blocks), dim3(NTH), shmem, stream,
                       x, la, b, c, d, out);
}